// GNN_H_50749333570143
// MI455X (gfx1250) — compile-verified
//
#include <hip/hip_runtime.h>

// MI455X / gfx1250. Key insight: pos/ghost are dead; x = embed[Z] has only 120
// distinct rows, so the 1M-node 4-layer MLP reduces exactly to
//   g_feat = sum_z count[z] * MLP(embed[z]).
// Remaining HBM traffic ~20 MB (Z histogram + w_g GEMV) -> ~1us at 23.3 TB/s.
// The dense 128x64 @ 64x64 MLP uses V_WMMA_F32_16X16X4_F32 (exact fp32).

typedef __attribute__((ext_vector_type(2))) float v2f;
typedef __attribute__((ext_vector_type(8))) float v8f;

#define HIDDEN   64
#define N_LAYERS 4
#define DIM_H    256
#define DIM_G    65536
#define N_Z      120
#define ZPAD     128

__device__ __forceinline__ float silu(float x) { return x / (1.0f + __expf(-x)); }

// ---------------- histogram of Z over 1M nodes ----------------
__global__ void zero_counts_kernel(unsigned* counts) {
    counts[threadIdx.x] = 0u;   // 128 threads
}

__global__ __launch_bounds__(256) void hist_kernel(const int* __restrict__ Z,
                                                   unsigned* __restrict__ counts, int n) {
    __shared__ unsigned h[ZPAD];
    for (int i = threadIdx.x; i < ZPAD; i += blockDim.x) h[i] = 0u;
    __syncthreads();
    int idx = blockIdx.x * blockDim.x + threadIdx.x;
    int stride = gridDim.x * blockDim.x;
    for (int i = idx; i < n; i += stride) {
        atomicAdd(&h[Z[i] & (ZPAD - 1)], 1u);
    }
    __syncthreads();
    for (int i = threadIdx.x; i < ZPAD; i += blockDim.x) {
        unsigned c = h[i];
        if (c) atomicAdd(&counts[i], c);
    }
}

// ------------- 4-layer MLP on the 120 distinct rows (WMMA f32) -------------
// One workgroup, 8 wave32s. State X is 128x64 f32 in LDS (rows 120..127 = 0;
// silu(0)=0 so padding stays 0 through all layers). Wave w owns M-tile w:
// it reads/writes only rows 16w..16w+15, so no intra-layer barrier is needed
// for X; barriers only protect the shared W staging buffer.
__global__ __launch_bounds__(256) void mlp_reduce_kernel(const float* __restrict__ embed,
                                                         const float* __restrict__ W_tp,
                                                         const unsigned* __restrict__ counts,
                                                         float* __restrict__ gfeat) {
    __shared__ float X[ZPAD][HIDDEN];        // 32 KB
    __shared__ float W[HIDDEN * HIDDEN];     // 16 KB
    const int tid = threadIdx.x;

    for (int i = tid; i < ZPAD * HIDDEN; i += 256) {
        int z = i / HIDDEN, hh = i % HIDDEN;
        X[z][hh] = (z < N_Z) ? embed[z * HIDDEN + hh] : 0.0f;
    }

    const int wave = tid >> 5;        // M-tile index 0..7
    const int lane = tid & 31;
    const int half = lane >> 4;       // 0: lanes 0-15, 1: lanes 16-31
    const int l16  = lane & 15;
    const int mrow = wave * 16 + l16; // A-matrix row owned by this lane

    for (int l = 0; l < N_LAYERS; ++l) {
        __syncthreads();
        for (int i = tid; i < HIDDEN * HIDDEN; i += 256)
            W[i] = W_tp[l * HIDDEN * HIDDEN + i];
        __syncthreads();

        float outbuf[4][8];
        for (int nt = 0; nt < 4; ++nt) {
            v8f acc = {0.f, 0.f, 0.f, 0.f, 0.f, 0.f, 0.f, 0.f};
            const int ncol = nt * 16 + l16;
            #pragma unroll
            for (int k0 = 0; k0 < HIDDEN; k0 += 4) {
                // A 16x4 f32: v0 = K=k0(+2*half), v1 = K=k0+1(+2*half)
                v2f a, b;
                const int ka = k0 + 2 * half;
                a.x = X[mrow][ka];
                a.y = X[mrow][ka + 1];
                // B 4x16 f32: v0 = rows k0 / k0+2, v1 = rows k0+1 / k0+3
                b.x = W[ka * HIDDEN + ncol];
                b.y = W[(ka + 1) * HIDDEN + ncol];
                acc = __builtin_amdgcn_wmma_f32_16x16x4_f32(
                    false, a, false, b, (short)0, acc, false, false);
            }
            #pragma unroll
            for (int r = 0; r < 8; ++r)
                outbuf[nt][r] = silu(acc[r] * 0.125f);   // INV_SQRT_H = 1/8
        }
        // D 16x16: VGPR r -> row r (lanes 0-15) / row r+8 (lanes 16-31)
        #pragma unroll
        for (int nt = 0; nt < 4; ++nt)
            #pragma unroll
            for (int r = 0; r < 8; ++r)
                X[wave * 16 + r + half * 8][nt * 16 + l16] = outbuf[nt][r];
    }
    __syncthreads();

    // g_feat[h] = sum_z count[z] * X[z][h]
    if (tid < HIDDEN) {
        float s = 0.0f;
        for (int z = 0; z < N_Z; ++z)
            s += (float)counts[z] * X[z][tid];
        gfeat[tid] = s;
    }
}

// ------------- h = g_feat @ w_h + b_h, symmetrized -> out[0..255] -------------
__global__ __launch_bounds__(256) void h_kernel(const float* __restrict__ gfeat,
                                                const float* __restrict__ w_h,
                                                const float* __restrict__ b_h,
                                                float* __restrict__ out) {
    __shared__ float hbuf[DIM_H];
    __shared__ float gf[HIDDEN];
    int tid = threadIdx.x;
    if (tid < HIDDEN) gf[tid] = gfeat[tid];
    __syncthreads();
    float s = b_h[tid];
    for (int k = 0; k < HIDDEN; ++k) s += gf[k] * w_h[k * DIM_H + tid];
    hbuf[tid] = s;
    __syncthreads();
    int i = tid >> 4, j = tid & 15;
    out[tid] = 0.5f * (hbuf[i * 16 + j] + hbuf[j * 16 + i]);
}

// ------------- g_raw = g_feat @ w_g + b_g (coalesced GEMV, 16 MB read) -------------
__global__ __launch_bounds__(256) void g_gemv_kernel(const float* __restrict__ gfeat,
                                                     const float* __restrict__ w_g,
                                                     const float* __restrict__ b_g,
                                                     float* __restrict__ graw) {
    __shared__ float gf[HIDDEN];
    int tid = threadIdx.x;
    if (tid < HIDDEN) gf[tid] = gfeat[tid];
    __syncthreads();
    int p = blockIdx.x * 256 + tid;
    float s = b_g[p];
    #pragma unroll 8
    for (int k = 0; k < HIDDEN; ++k)
        s += gf[k] * w_g[(size_t)k * DIM_G + p];
    graw[p] = s;
}

// ------------- 8-fold permutation average (group => gather is exact) -------------
__global__ __launch_bounds__(256) void g_sym_kernel(const float* __restrict__ graw,
                                                    float* __restrict__ out) {
    int p = blockIdx.x * 256 + threadIdx.x;
    int i = (p >> 12) & 15, j = (p >> 8) & 15, k = (p >> 4) & 15, l = p & 15;
#define G4(a, b, c, d) graw[((((a) * 16 + (b)) * 16 + (c)) * 16) + (d)]
    float s = G4(i, j, k, l) + G4(j, i, k, l) + G4(i, j, l, k) + G4(j, i, l, k)
            + G4(k, l, i, j) + G4(k, l, j, i) + G4(l, k, i, j) + G4(l, k, j, i);
#undef G4
    out[DIM_H + p] = 0.125f * s;
}

extern "C" void kernel_launch(void* const* d_in, const int* in_sizes, int n_in,
                              void* d_out, int out_size, void* d_ws, size_t ws_size,
                              hipStream_t stream) {
    const int*   Z     = (const int*)  d_in[0];
    // d_in[1] = pos   (dead input)
    // d_in[2] = ghost (dead input)
    const float* embed = (const float*)d_in[3];
    const float* W_tp  = (const float*)d_in[4];
    const float* w_h   = (const float*)d_in[5];
    const float* b_h   = (const float*)d_in[6];
    const float* w_g   = (const float*)d_in[7];
    const float* b_g   = (const float*)d_in[8];
    float* out = (float*)d_out;

    char* ws = (char*)d_ws;
    unsigned* counts = (unsigned*)ws;          // 128 x u32
    float*    gfeat  = (float*)(ws + 512);     // 64 x f32
    float*    graw   = (float*)(ws + 1024);    // 65536 x f32 (256 KB)

    int n = in_sizes[0];

    zero_counts_kernel<<<1, 128, 0, stream>>>(counts);
    hist_kernel<<<1024, 256, 0, stream>>>(Z, counts, n);
    mlp_reduce_kernel<<<1, 256, 0, stream>>>(embed, W_tp, counts, gfeat);
    h_kernel<<<1, 256, 0, stream>>>(gfeat, w_h, b_h, out);
    g_gemv_kernel<<<DIM_G / 256, 256, 0, stream>>>(gfeat, w_g, b_g, graw);
    g_sym_kernel<<<DIM_G / 256, 256, 0, stream>>>(graw, out);
}